// Gnn_83562883711168
// MI455X (gfx1250) — compile-verified
//
#include <hip/hip_runtime.h>
#include <hip/hip_bf16.h>

typedef __attribute__((ext_vector_type(2))) float v2f;
typedef __attribute__((ext_vector_type(8))) float v8f;

#define LN_EPS 1e-5f

// ---------------------------------------------------------------------------
// Native f32 atomic add (global_atomic_add_f32) — avoid CAS loop fallback.
// ---------------------------------------------------------------------------
__device__ __forceinline__ void atomic_add_f32(float* p, float v) {
    unsafeAtomicAdd(p, v);
}

// ---------------------------------------------------------------------------
// Degree: deg[i] = 1 (self loop) + #incoming edges
// ---------------------------------------------------------------------------
__global__ void deg_init_kernel(float* __restrict__ deg, int n) {
    int i = blockIdx.x * blockDim.x + threadIdx.x;
    if (i < n) deg[i] = 1.0f;
}

__global__ void deg_acc_kernel(const int* __restrict__ dst, float* __restrict__ deg, int e) {
    int i = blockIdx.x * blockDim.x + threadIdx.x;
    if (i < e) atomic_add_f32(&deg[dst[i]], 1.0f);
}

__global__ void dinv_kernel(const float* __restrict__ deg, float* __restrict__ dinv, int n) {
    int i = blockIdx.x * blockDim.x + threadIdx.x;
    if (i < n) {
        float d = deg[i];
        dinv[i] = (d > 0.0f) ? rsqrtf(d) : 0.0f;
    }
}

// ---------------------------------------------------------------------------
// GEMM H[N x 128] = X[N x 128] * W[128 x 128] via V_WMMA_F32_16X16X4_F32.
// Block = 256 threads = 8 waves. Wave w computes the 16x16 tile
// (mtile = blockIdx.x, ntile = w). K stepped by 4 -> 32 WMMA per tile.
//
// A 16x4 layout : lane%16 = M, VGPR j holds K = k0 + 2*(lane/16) + j
// B 4x16 layout : lane%16 = N, VGPR j holds K = k0 + 2*(lane/16) + j
// C/D 16x16     : VGPR r holds M = r + 8*(lane/16), N = lane%16
// EXEC all-ones guaranteed: N % 16 == 0, no divergence.
// ---------------------------------------------------------------------------
__global__ void gemm_wmma_kernel(const float* __restrict__ X,
                                 const float* __restrict__ W,
                                 float* __restrict__ H) {
    const int wave = threadIdx.x >> 5;      // 0..7 -> ntile
    const int lane = threadIdx.x & 31;
    const int l    = lane & 15;
    const int hi   = lane >> 4;             // 0 or 1
    const int mtile = blockIdx.x;

    const int arow = mtile * 16 + l;        // A row (M) for this lane
    const int ncol = wave * 16 + l;         // B/D column (N) for this lane

    const float* __restrict__ xrow = X + (size_t)arow * 128;

    v8f c = {};
#pragma unroll
    for (int k0 = 0; k0 < 128; k0 += 4) {
        const int ka = k0 + 2 * hi;
        v2f a = *(const v2f*)(xrow + ka);                 // contiguous K pair
        v2f b;
        b.x = W[(size_t)ka * 128 + ncol];
        b.y = W[(size_t)(ka + 1) * 128 + ncol];
        c = __builtin_amdgcn_wmma_f32_16x16x4_f32(
                /*neg_a=*/false, a, /*neg_b=*/false, b,
                /*c_mod=*/(short)0, c, /*reuse_a=*/false, /*reuse_b=*/false);
    }

#pragma unroll
    for (int r = 0; r < 8; ++r) {
        H[(size_t)(mtile * 16 + 8 * hi + r) * 128 + ncol] = c[r];
    }
}

// ---------------------------------------------------------------------------
// Out[i] = bias + H[i] * dinv[i]^2   (fused bias + self-loop message)
// One thread per float4 => 32 threads per row.
// ---------------------------------------------------------------------------
__global__ void init_out_kernel(const float* __restrict__ H,
                                const float* __restrict__ bias,
                                const float* __restrict__ dinv,
                                float* __restrict__ Out, int nrows) {
    int idx = blockIdx.x * blockDim.x + threadIdx.x;
    int row = idx >> 5;
    int c4  = idx & 31;
    if (row >= nrows) return;
    float di = dinv[row];
    float w  = di * di;
    const float4 v  = *(const float4*)(H + (size_t)row * 128 + c4 * 4);
    const float4 bb = *(const float4*)(bias + c4 * 4);
    float4 o;
    o.x = fmaf(v.x, w, bb.x);
    o.y = fmaf(v.y, w, bb.y);
    o.z = fmaf(v.z, w, bb.z);
    o.w = fmaf(v.w, w, bb.w);
    *(float4*)(Out + (size_t)row * 128 + c4 * 4) = o;
}

// ---------------------------------------------------------------------------
// Edge scatter: one wave per edge; lane handles 4 features.
// Out[dst] += H[src] * (dinv[src]*dinv[dst]).  H & Out both fit in L2.
// ---------------------------------------------------------------------------
__global__ void scatter_kernel(const float* __restrict__ H,
                               const int* __restrict__ src,
                               const int* __restrict__ dst,
                               const float* __restrict__ dinv,
                               float* __restrict__ Out, int nedges) {
    int wid  = (blockIdx.x * blockDim.x + threadIdx.x) >> 5;
    int lane = threadIdx.x & 31;
    if (wid >= nedges) return;
    int s = src[wid];
    int d = dst[wid];
    float norm = dinv[s] * dinv[d];
    const float4 v = *(const float4*)(H + (size_t)s * 128 + lane * 4);
    float* o = Out + (size_t)d * 128 + lane * 4;
    atomic_add_f32(o + 0, v.x * norm);
    atomic_add_f32(o + 1, v.y * norm);
    atomic_add_f32(o + 2, v.z * norm);
    atomic_add_f32(o + 3, v.w * norm);
}

// ---------------------------------------------------------------------------
// LayerNorm(feature dim) * gamma + beta, then ReLU. One wave32 per row;
// lane holds a float4; butterfly reduction with __shfl_xor (warpSize == 32).
// Safe in place (each wave reads then writes only its own row).
// ---------------------------------------------------------------------------
__global__ void ln_relu_kernel(const float* __restrict__ In,
                               float* __restrict__ Out,
                               const float* __restrict__ gamma,
                               const float* __restrict__ beta, int nrows) {
    int wid  = (blockIdx.x * blockDim.x + threadIdx.x) >> 5;
    int lane = threadIdx.x & 31;
    if (wid >= nrows) return;

    const float4 v = *(const float4*)(In + (size_t)wid * 128 + lane * 4);
    float s  = v.x + v.y + v.z + v.w;
    float sq = v.x * v.x + v.y * v.y + v.z * v.z + v.w * v.w;
#pragma unroll
    for (int m = 16; m >= 1; m >>= 1) {
        s  += __shfl_xor(s,  m, 32);
        sq += __shfl_xor(sq, m, 32);
    }
    const float mean = s * (1.0f / 128.0f);
    const float var  = fmaxf(sq * (1.0f / 128.0f) - mean * mean, 0.0f);
    const float rstd = rsqrtf(var + LN_EPS);

    const float4 g = *(const float4*)(gamma + lane * 4);
    const float4 b = *(const float4*)(beta  + lane * 4);
    float4 o;
    o.x = fmaxf((v.x - mean) * rstd * g.x + b.x, 0.0f);
    o.y = fmaxf((v.y - mean) * rstd * g.y + b.y, 0.0f);
    o.z = fmaxf((v.z - mean) * rstd * g.z + b.z, 0.0f);
    o.w = fmaxf((v.w - mean) * rstd * g.w + b.w, 0.0f);
    *(float4*)(Out + (size_t)wid * 128 + lane * 4) = o;
}

// ---------------------------------------------------------------------------
// Orchestration
// ---------------------------------------------------------------------------
extern "C" void kernel_launch(void* const* d_in, const int* in_sizes, int n_in,
                              void* d_out, int out_size, void* d_ws, size_t ws_size,
                              hipStream_t stream) {
    const float* x   = (const float*)d_in[0];
    const int*   ei  = (const int*)d_in[1];
    const float* W1  = (const float*)d_in[2];
    const float* b1  = (const float*)d_in[3];
    const float* g1  = (const float*)d_in[4];
    const float* be1 = (const float*)d_in[5];
    const float* W2  = (const float*)d_in[6];
    const float* b2  = (const float*)d_in[7];
    const float* g2  = (const float*)d_in[8];
    const float* be2 = (const float*)d_in[9];

    const int N = in_sizes[0] / 128;   // 100000, divisible by 16
    const int E = in_sizes[1] / 2;     // 1600000

    const int* src = ei;
    const int* dst = ei + E;

    float* ws   = (float*)d_ws;
    float* deg  = ws;                        // N floats
    float* dinv = ws + (size_t)N;            // N floats
    float* A    = ws + 2 * (size_t)N;        // N*128 floats
    float* B    = A + (size_t)N * 128;       // N*128 floats
    float* out  = (float*)d_out;

    const int T = 256;

    // Symmetric-norm degrees (shared by both layers).
    deg_init_kernel<<<(N + T - 1) / T, T, 0, stream>>>(deg, N);
    deg_acc_kernel<<<(E + T - 1) / T, T, 0, stream>>>(dst, deg, E);
    dinv_kernel<<<(N + T - 1) / T, T, 0, stream>>>(deg, dinv, N);

    const int gemm_blocks    = N / 16;                 // one 16-row stripe / block
    const int elem4_blocks   = (N * 32 + T - 1) / T;   // float4 per thread
    const int edge_blocks    = (E + 7) / 8;            // one wave per edge
    const int row_blocks     = (N + 7) / 8;            // one wave per row

    // ---- Layer 1 ----
    gemm_wmma_kernel<<<gemm_blocks, T, 0, stream>>>(x, W1, A);
    init_out_kernel<<<elem4_blocks, T, 0, stream>>>(A, b1, dinv, B, N);
    scatter_kernel<<<edge_blocks, T, 0, stream>>>(A, src, dst, dinv, B, E);
    ln_relu_kernel<<<row_blocks, T, 0, stream>>>(B, A, g1, be1, N);   // A := act1

    // ---- Layer 2 ----
    gemm_wmma_kernel<<<gemm_blocks, T, 0, stream>>>(A, W2, B);        // B := h2
    init_out_kernel<<<elem4_blocks, T, 0, stream>>>(B, b2, dinv, out, N);
    scatter_kernel<<<edge_blocks, T, 0, stream>>>(B, src, dst, dinv, out, E);
    ln_relu_kernel<<<row_blocks, T, 0, stream>>>(out, out, g2, be2, N); // in place
}